// IndexedLinearLayer_81870666597189
// MI455X (gfx1250) — compile-verified
//
#include <hip/hip_runtime.h>

#define SIZE_IN   8192
#define SIZE_OUT  2048

typedef __attribute__((ext_vector_type(2))) float v2f;
typedef __attribute__((ext_vector_type(8))) float v8f;

// ---------------------------------------------------------------------------
// Pass 1: streaming gather-GEMV partial sums.
// grid = (nchunks, 2); block = 256 threads; each thread owns 4 columns (float4).
// Block (c, t) accumulates features [c*fpc, (c+1)*fpc) for columns
// [t*1024, t*1024+1024) and writes fp32 partials to ws[c][col].
// Memory-bound (0.5 FLOP/byte): one fmaf per loaded float; per feature each
// block issues one fully coalesced 4 KB load (256 lanes x 16 B) from the
// gathered row, with a global_prefetch_b8 running 8 features ahead.
// ---------------------------------------------------------------------------
__global__ void __launch_bounds__(256)
idxlin_partial(const float* __restrict__ x,
               const int*   __restrict__ idx,
               const float* __restrict__ W,
               float*       __restrict__ partial,
               int feats_per_chunk) {
  const int chunk = blockIdx.x;
  const int col   = blockIdx.y * 1024 + threadIdx.x * 4;
  const int f0    = chunk * feats_per_chunk;

  float ax = 0.f, ay = 0.f, az = 0.f, aw = 0.f;

#pragma unroll 4
  for (int f = 0; f < feats_per_chunk; ++f) {
    const int ff = f0 + f;
    const float     xs = x[ff];                        // uniform -> scalar load
    const long long q  = (long long)idx[ff];           // uniform -> scalar load
    const float* row = W + ((size_t)q * SIZE_IN + (size_t)ff) * SIZE_OUT + col;

    // Prefetch the gathered row 8 features ahead (global_prefetch_b8).
    if (f + 8 < feats_per_chunk) {
      const int fp = ff + 8;
      const float* prow =
          W + ((size_t)idx[fp] * SIZE_IN + (size_t)fp) * SIZE_OUT + col;
      __builtin_prefetch(prow, 0, 0);
    }

    const float4 w = *(const float4*)row;
    ax = fmaf(xs, w.x, ax);
    ay = fmaf(xs, w.y, ay);
    az = fmaf(xs, w.z, az);
    aw = fmaf(xs, w.w, aw);
  }

  float4* dst = (float4*)(partial + (size_t)chunk * SIZE_OUT + col);
  *dst = make_float4(ax, ay, az, aw);
}

// ---------------------------------------------------------------------------
// Pass 2: reduce NCHUNKS partials per column on the matrix unit.
// A = all-ones 16x4  =>  D[m,n] = C[m,n] + sum_k B[k,n].
// Per wave: 16 columns. Lane L (L<16) carries chunks {4t, 4t+1} for column
// n0+L in its two B VGPRs; lane L+16 carries chunks {4t+2, 4t+3}. Since A is
// all ones, the sum over K is symmetric, so the result is correct for ANY
// (K,lane) assignment of the B layout. Result taken from D VGPR0 (M=0 row),
// lane = N (documented C/D layout).
// NCHUNKS is a template param (always a multiple of 4) so the loop has NO
// bounds guards: unconditional loads, EXEC stays all-ones, full unroll.
// ---------------------------------------------------------------------------
template <int NCHUNKS>
__global__ void __launch_bounds__(256)
idxlin_reduce_wmma(const float* __restrict__ partial,
                   const float* __restrict__ bias,
                   float*       __restrict__ out) {
  const int lane = threadIdx.x & 31;
  const int wave = (blockIdx.x * blockDim.x + threadIdx.x) >> 5;
  const int n    = wave * 16 + (lane & 15);   // output column for this lane
  const int half = (lane >> 4) & 1;           // 0: chunks 4t,4t+1  1: 4t+2,4t+3

  const float* base = partial + (size_t)(2 * half) * SIZE_OUT + n;

  v2f a; a[0] = 1.0f; a[1] = 1.0f;
  v8f c = {};

#pragma unroll
  for (int t = 0; t < NCHUNKS / 4; ++t) {
    v2f b;
    b[0] = base[(size_t)(4 * t)     * SIZE_OUT];
    b[1] = base[(size_t)(4 * t + 1) * SIZE_OUT];
    // v_wmma_f32_16x16x4_f32: (neg_a, A, neg_b, B, c_mod, C, reuse_a, reuse_b)
    c = __builtin_amdgcn_wmma_f32_16x16x4_f32(false, a, false, b,
                                              (short)0, c, false, false);
  }

  if (lane < 16) out[n] = c[0] + bias[n];
}

// ---------------------------------------------------------------------------
// Fallback (only if d_ws is unusably small): direct single-pass reduction.
// ---------------------------------------------------------------------------
__global__ void __launch_bounds__(256)
idxlin_direct(const float* __restrict__ x,
              const int*   __restrict__ idx,
              const float* __restrict__ W,
              const float* __restrict__ bias,
              float*       __restrict__ out) {
  const int col = blockIdx.x * blockDim.x + threadIdx.x;  // 0..2047
  float acc = 0.f;
  for (int f = 0; f < SIZE_IN; ++f) {
    const long long q = (long long)idx[f];
    acc = fmaf(x[f], W[((size_t)q * SIZE_IN + (size_t)f) * SIZE_OUT + col], acc);
  }
  out[col] = acc + bias[col];
}

extern "C" void kernel_launch(void* const* d_in, const int* in_sizes, int n_in,
                              void* d_out, int out_size, void* d_ws, size_t ws_size,
                              hipStream_t stream) {
  const float* x    = (const float*)d_in[0];
  const int*   idx  = (const int*)  d_in[1];
  const float* W    = (const float*)d_in[2];
  const float* bias = (const float*)d_in[3];
  float*       out  = (float*)d_out;
  float*       ws   = (float*)d_ws;

  const size_t chunk_bytes = (size_t)SIZE_OUT * sizeof(float);

  if (ws_size >= 64 * chunk_bytes) {
    dim3 grid(64, 2);  // 64 feature chunks x 2 column tiles (1024 cols each)
    idxlin_partial<<<grid, 256, 0, stream>>>(x, idx, W, ws, SIZE_IN / 64);
    // 2048 cols / 16 per wave = 128 waves = 16 blocks x 8 waves
    idxlin_reduce_wmma<64><<<16, 256, 0, stream>>>(ws, bias, out);
  } else if (ws_size >= 16 * chunk_bytes) {
    dim3 grid(16, 2);
    idxlin_partial<<<grid, 256, 0, stream>>>(x, idx, W, ws, SIZE_IN / 16);
    idxlin_reduce_wmma<16><<<16, 256, 0, stream>>>(ws, bias, out);
  } else if (ws_size >= 4 * chunk_bytes) {
    dim3 grid(4, 2);
    idxlin_partial<<<grid, 256, 0, stream>>>(x, idx, W, ws, SIZE_IN / 4);
    idxlin_reduce_wmma<4><<<16, 256, 0, stream>>>(ws, bias, out);
  } else {
    idxlin_direct<<<SIZE_OUT / 256, 256, 0, stream>>>(x, idx, W, bias, out);
  }
}